// RecurrentGCN_41686952575128
// MI455X (gfx1250) — compile-verified
//
#include <hip/hip_runtime.h>
#include <math.h>

typedef float v2f __attribute__((ext_vector_type(2)));
typedef float v8f __attribute__((ext_vector_type(8)));

#define WMMA4(a, b, c) \
  __builtin_amdgcn_wmma_f32_16x16x4_f32(false, (a), false, (b), (short)0, (c), false, false)

// ---- ordered-bits encoding for float atomicMax on unsigned ----
__device__ __forceinline__ unsigned fenc(float f) {
  unsigned b = __float_as_uint(f);
  return (b & 0x80000000u) ? ~b : (b | 0x80000000u);
}
__device__ __forceinline__ float fdec(unsigned u) {
  return (u & 0x80000000u) ? __uint_as_float(u & 0x7FFFFFFFu) : __uint_as_float(~u);
}

// ============================ init =====================================
__global__ void k_init_amax(unsigned* amax, int N) {
  int n = blockIdx.x * blockDim.x + threadIdx.x;
  if (n < N) amax[n] = fenc(-INFINITY);
}

// =================== GAT: degree count + edge_attr mean =================
__global__ void k_cnt_easum(const int* ei, const float* ea, float* cnt, float* eam, int E) {
  int tid = blockIdx.x * blockDim.x + threadIdx.x;
  if (tid >= E * 16) return;
  int e = tid >> 4, f = tid & 15;
  int d = ei[E + e];
  if (f == 0) atomicAdd(&cnt[d], 1.0f);
  atomicAdd(&eam[d * 16 + f], ea[e * 16 + f]);
}

__global__ void k_eamean(float* eam, const float* cnt, int N) {
  int tid = blockIdx.x * blockDim.x + threadIdx.x;
  if (tid >= N * 16) return;
  int n = tid >> 4;
  eam[tid] /= fmaxf(cnt[n], 1.0f);
}

// ============ WMMA GEMM: xl = x@Wl + bl ; xr = x@Wr + br ================
// one wave32 per 16-row tile; K = 128 in steps of 4 (f32 WMMA), 64 wmma/tile
__global__ void k_gemm_xlxr(const float* __restrict__ x,
                            const float* __restrict__ Wl, const float* __restrict__ bl,
                            const float* __restrict__ Wr, const float* __restrict__ br,
                            float* __restrict__ xl, float* __restrict__ xr) {
  int tile = blockIdx.x;
  int lane = threadIdx.x;          // 0..31, all active: EXEC all-1s for WMMA
  int half = lane >> 4;            // 0: K={0,1}, 1: K={2,3}
  int l16 = lane & 15;
  int row = tile * 16 + l16;       // A-matrix row held by this lane
  v8f acc0 = {}; v8f acc1 = {};
#pragma unroll
  for (int k = 0; k < 128; k += 4) {
    int ka = k + half * 2;
    v2f a;  a.x = x[row * 128 + ka];     a.y = x[row * 128 + ka + 1];
    v2f b0; b0.x = Wl[ka * 16 + l16];    b0.y = Wl[(ka + 1) * 16 + l16];
    v2f b1; b1.x = Wr[ka * 16 + l16];    b1.y = Wr[(ka + 1) * 16 + l16];
    acc0 = WMMA4(a, b0, acc0);
    acc1 = WMMA4(a, b1, acc1);
  }
#pragma unroll
  for (int i = 0; i < 8; ++i) {
    int m = tile * 16 + i + half * 8;    // D: VGPR i -> row i (+8 upper lanes)
    int c = l16;
    xl[m * 16 + c] = acc0[i] + bl[c];
    xr[m * 16 + c] = acc1[i] + br[c];
  }
}

// ================= GAT attention logits (real edges) ====================
__global__ void k_edge_alpha(const int* ei, const float* __restrict__ ea,
                             const float* __restrict__ xl, const float* __restrict__ xr,
                             const float* We, const float* att,
                             float* aedge, unsigned* amax, int E) {
  __shared__ float sWe[256];
  __shared__ float satt[16];
  int t = threadIdx.x;
  sWe[t] = We[t];
  if (t < 16) satt[t] = att[t];
  __syncthreads();
  int e = blockIdx.x * blockDim.x + t;
  if (e >= E) return;
  int s = ei[e], d = ei[E + e];
  float eav[16];
#pragma unroll
  for (int j = 0; j < 16; ++j) eav[j] = ea[e * 16 + j];
  float acc = 0.0f;
#pragma unroll
  for (int f = 0; f < 16; ++f) {
    float m = xl[s * 16 + f] + xr[d * 16 + f];
#pragma unroll
    for (int j = 0; j < 16; ++j) m += eav[j] * sWe[j * 16 + f];
    m = m > 0.0f ? m : 0.2f * m;     // leaky_relu(0.2)
    acc += m * satt[f];
  }
  aedge[e] = acc;
  atomicMax(&amax[d], fenc(acc));
}

// ============== GAT attention logits (self loops, em = ea_mean@We) ======
__global__ void k_loop_alpha(const float* __restrict__ eam,
                             const float* __restrict__ xl, const float* __restrict__ xr,
                             const float* We, const float* att,
                             float* aloop, unsigned* amax, int N) {
  __shared__ float sWe[256];
  __shared__ float satt[16];
  int t = threadIdx.x;
  sWe[t] = We[t];
  if (t < 16) satt[t] = att[t];
  __syncthreads();
  int n = blockIdx.x * blockDim.x + t;
  if (n >= N) return;
  float eav[16];
#pragma unroll
  for (int j = 0; j < 16; ++j) eav[j] = eam[n * 16 + j];
  float acc = 0.0f;
#pragma unroll
  for (int f = 0; f < 16; ++f) {
    float m = xl[n * 16 + f] + xr[n * 16 + f];
#pragma unroll
    for (int j = 0; j < 16; ++j) m += eav[j] * sWe[j * 16 + f];
    m = m > 0.0f ? m : 0.2f * m;
    acc += m * satt[f];
  }
  aloop[n] = acc;
  atomicMax(&amax[n], fenc(acc));
}

// ==================== softmax: exp + denominator ========================
__global__ void k_exp_edge(const int* ei, float* aedge, const unsigned* amax,
                           float* denom, int E) {
  int e = blockIdx.x * blockDim.x + threadIdx.x;
  if (e >= E) return;
  int d = ei[E + e];
  float ex = expf(aedge[e] - fdec(amax[d]));
  aedge[e] = ex;
  atomicAdd(&denom[d], ex);
}

__global__ void k_exp_loop(float* aloop, const unsigned* amax, float* denom, int N) {
  int n = blockIdx.x * blockDim.x + threadIdx.x;
  if (n >= N) return;
  float ex = expf(aloop[n] - fdec(amax[n]));
  aloop[n] = ex;
  atomicAdd(&denom[n], ex);
}

// ==================== weighted scatter of xl[src] =======================
__global__ void k_scatter(const int* ei, const float* __restrict__ aedge,
                          const float* __restrict__ denom, const float* __restrict__ xl,
                          float* hsum, int E) {
  int tid = blockIdx.x * blockDim.x + threadIdx.x;
  if (tid >= E * 16) return;
  int e = tid >> 4, f = tid & 15;
  int s = ei[e], d = ei[E + e];
  float w = aedge[e] / (denom[d] + 1e-16f);
  atomicAdd(&hsum[d * 16 + f], w * xl[s * 16 + f]);
}

// self-loop contribution + gat bias, in place: hsum -> h
__global__ void k_gatfin(float* hsum, const float* aloop, const float* denom,
                         const float* xl, const float* gbias, int N) {
  int tid = blockIdx.x * blockDim.x + threadIdx.x;
  if (tid >= N * 16) return;
  int n = tid >> 4, f = tid & 15;
  float w = aloop[n] / (denom[n] + 1e-16f);
  hsum[tid] = hsum[tid] + w * xl[n * 16 + f] + gbias[f];
}

// ================ Cheb: degree (over src!) and dinv =====================
__global__ void k_deg(const int* ei, float* deg, int E) {
  int e = blockIdx.x * blockDim.x + threadIdx.x;
  if (e >= E) return;
  int s = ei[e], d = ei[E + e];
  if (s != d) atomicAdd(&deg[s], 1.0f);
}

__global__ void k_dinv(float* deg, int N) {
  int n = blockIdx.x * blockDim.x + threadIdx.x;
  if (n >= N) return;
  float g = deg[n];
  deg[n] = (g > 0.0f) ? rsqrtf(fmaxf(g, 1e-12f)) : 0.0f;
}

// sparse matvec: out[dst] += -dinv[src]*dinv[dst]*in[src]  (src!=dst)
__global__ void k_matvec(const int* ei, const float* __restrict__ dinv,
                         const float* __restrict__ in, float* out, int E) {
  int tid = blockIdx.x * blockDim.x + threadIdx.x;
  if (tid >= E * 16) return;
  int e = tid >> 4, f = tid & 15;
  int s = ei[e], d = ei[E + e];
  if (s == d) return;
  float nrm = -dinv[s] * dinv[d];
  atomicAdd(&out[d * 16 + f], nrm * in[s * 16 + f]);
}

// T = 2*T - Tprev (Chebyshev recurrence)
__global__ void k_chebrec(float* T, const float* Tprev, int N) {
  int tid = blockIdx.x * blockDim.x + threadIdx.x;
  if (tid >= N * 16) return;
  T[tid] = 2.0f * T[tid] - Tprev[tid];
}

// ====== fused GRU GEMM: [T0|T1|T2|T3] (Nx64) @ {Wz,Wh} (64x16), WMMA ====
// then Z = sigmoid(.+b0+b1), Ht = tanh(.+b4+b5), H = (1-Z)*Ht
__global__ void k_gru(const float* __restrict__ T0, const float* __restrict__ T1,
                      const float* __restrict__ T2, const float* __restrict__ T3,
                      const float* __restrict__ chebW, const float* __restrict__ chebB,
                      float* __restrict__ H) {
  int tile = blockIdx.x;
  int lane = threadIdx.x;
  int half = lane >> 4;
  int l16 = lane & 15;
  int row = tile * 16 + l16;
  v8f accZ = {}; v8f accH = {};
#pragma unroll
  for (int t = 0; t < 4; ++t) {
    const float* Tm = (t == 0) ? T0 : (t == 1) ? T1 : (t == 2) ? T2 : T3;
    const float* Wz = chebW + t * 256;          // cheb_W[0][t]
    const float* Wh = chebW + (16 + t) * 256;   // cheb_W[4][t]
#pragma unroll
    for (int kc = 0; kc < 16; kc += 4) {
      int ka = kc + half * 2;
      v2f a;  a.x = Tm[row * 16 + ka];      a.y = Tm[row * 16 + ka + 1];
      v2f bz; bz.x = Wz[ka * 16 + l16];     bz.y = Wz[(ka + 1) * 16 + l16];
      v2f bh; bh.x = Wh[ka * 16 + l16];     bh.y = Wh[(ka + 1) * 16 + l16];
      accZ = WMMA4(a, bz, accZ);
      accH = WMMA4(a, bh, accH);
    }
  }
#pragma unroll
  for (int i = 0; i < 8; ++i) {
    int m = tile * 16 + i + half * 8;
    int c = l16;
    float zp = accZ[i] + chebB[c] + chebB[16 + c];        // b[0]+b[1]
    float hp = accH[i] + chebB[64 + c] + chebB[80 + c];   // b[4]+b[5]
    float z = 1.0f / (1.0f + expf(-zp));
    float ht = tanhf(hp);
    H[m * 16 + c] = (1.0f - z) * ht;                      // H0=0 => Z*H0 drops
  }
}

// =================== per-node MLP 16 -> 4 -> 8 ==========================
// idx 0 -> all-zero input row; idx>=1 -> H[idx-1] (only W1 rows 0..15 matter)
__global__ void k_mlp(const float* __restrict__ H, const float* W1, const float* b1,
                      const float* W2, const float* b2, float* out, int N) {
  int idx = blockIdx.x * blockDim.x + threadIdx.x;
  if (idx > N) return;
  float hid[4];
#pragma unroll
  for (int j = 0; j < 4; ++j) {
    float s = b1[j];
    if (idx > 0) {
      const float* hr = H + (size_t)(idx - 1) * 16;
#pragma unroll
      for (int f = 0; f < 16; ++f) s += hr[f] * W1[f * 4 + j];
    }
    hid[j] = fmaxf(s, 0.0f);
  }
#pragma unroll
  for (int o = 0; o < 8; ++o) {
    float s = b2[o];
#pragma unroll
    for (int j = 0; j < 4; ++j) s += hid[j] * W2[j * 8 + o];
    out[idx * 8 + o] = s;
  }
}

// ========================================================================
extern "C" void kernel_launch(void* const* d_in, const int* in_sizes, int n_in,
                              void* d_out, int out_size, void* d_ws, size_t ws_size,
                              hipStream_t stream) {
  const float* x     = (const float*)d_in[0];
  const int*   ei    = (const int*)  d_in[1];
  const float* ea    = (const float*)d_in[2];
  const float* Wl    = (const float*)d_in[3];
  const float* bl    = (const float*)d_in[4];
  const float* Wr    = (const float*)d_in[5];
  const float* br    = (const float*)d_in[6];
  const float* We    = (const float*)d_in[7];
  const float* att   = (const float*)d_in[8];
  const float* gbias = (const float*)d_in[9];
  const float* chebW = (const float*)d_in[10];
  const float* chebB = (const float*)d_in[11];
  const float* W1    = (const float*)d_in[12];
  const float* b1    = (const float*)d_in[13];
  const float* W2    = (const float*)d_in[14];
  const float* b2    = (const float*)d_in[15];
  float* out = (float*)d_out;

  const int N = in_sizes[0] / 128;   // 50000
  const int E = in_sizes[1] / 2;     // 1600000
  const size_t NF = (size_t)N * 16;

  // workspace layout (floats)
  float* ws    = (float*)d_ws;
  float* xl    = ws;                 // N*16   (later reused to hold GRU H)
  float* xr    = xl + NF;            // N*16
  float* eam   = xr + NF;            // N*16  easum -> ea_mean
  float* hsum  = eam + NF;           // N*16  GAT accum -> h
  float* T1    = hsum + NF;          // N*16
  float* T2    = T1 + NF;            // N*16
  float* T3    = T2 + NF;            // N*16
  float* cnt   = T3 + NF;            // N
  float* denom = cnt + N;            // N
  float* aloop = denom + N;          // N
  float* dinv  = aloop + N;          // N  (deg -> dinv in place)
  unsigned* amax = (unsigned*)(dinv + N);  // N
  float* aedge = (float*)(amax + N); // E  (alpha -> exp(alpha-max) in place)

  const int TPB = 256;
  int gE   = (E + TPB - 1) / TPB;
  int gEF  = (E * 16 + TPB - 1) / TPB;
  int gN   = (N + TPB - 1) / TPB;
  int gNF  = (N * 16 + TPB - 1) / TPB;
  int gN1  = (N + 1 + TPB - 1) / TPB;
  int tiles = N / 16;                // 3125 (N divisible by 16)

  // zero accumulators (eam..T3 contiguous), counters
  hipMemsetAsync(eam, 0, NF * 5 * sizeof(float), stream);
  hipMemsetAsync(cnt, 0, (size_t)N * 2 * sizeof(float), stream);
  hipMemsetAsync(dinv, 0, (size_t)N * sizeof(float), stream);
  k_init_amax<<<gN, TPB, 0, stream>>>(amax, N);

  // ---- GATv2 ----
  k_cnt_easum<<<gEF, TPB, 0, stream>>>(ei, ea, cnt, eam, E);
  k_eamean<<<gNF, TPB, 0, stream>>>(eam, cnt, N);
  k_gemm_xlxr<<<tiles, 32, 0, stream>>>(x, Wl, bl, Wr, br, xl, xr);   // WMMA f32
  k_edge_alpha<<<gE, TPB, 0, stream>>>(ei, ea, xl, xr, We, att, aedge, amax, E);
  k_loop_alpha<<<gN, TPB, 0, stream>>>(eam, xl, xr, We, att, aloop, amax, N);
  k_exp_edge<<<gE, TPB, 0, stream>>>(ei, aedge, amax, denom, E);
  k_exp_loop<<<gN, TPB, 0, stream>>>(aloop, amax, denom, N);
  k_scatter<<<gEF, TPB, 0, stream>>>(ei, aedge, denom, xl, hsum, E);
  k_gatfin<<<gNF, TPB, 0, stream>>>(hsum, aloop, denom, xl, gbias, N); // hsum == h

  // ---- Chebyshev normalization ----
  k_deg<<<gE, TPB, 0, stream>>>(ei, dinv, E);
  k_dinv<<<gN, TPB, 0, stream>>>(dinv, N);

  // ---- Chebyshev basis T1..T3 (T0 = h) ----
  k_matvec<<<gEF, TPB, 0, stream>>>(ei, dinv, hsum, T1, E);
  k_matvec<<<gEF, TPB, 0, stream>>>(ei, dinv, T1, T2, E);
  k_chebrec<<<gNF, TPB, 0, stream>>>(T2, hsum, N);
  k_matvec<<<gEF, TPB, 0, stream>>>(ei, dinv, T2, T3, E);
  k_chebrec<<<gNF, TPB, 0, stream>>>(T3, T1, N);

  // ---- fused GRU GEMM (WMMA) -> H (stored in xl, free after GAT) ----
  k_gru<<<tiles, 32, 0, stream>>>(hsum, T1, T2, T3, chebW, chebB, xl);

  // ---- per-node MLP -> out[(N+1) x 8] ----
  k_mlp<<<gN1, TPB, 0, stream>>>(xl, W1, b1, W2, b2, out, N);
}